// INT4AsymmetricWeightsDecompressor_81673098101539
// MI455X (gfx1250) — compile-verified
//
#include <hip/hip_runtime.h>

// INT4 asymmetric weight dequant:  out = (fp16)( (w4 - zp4) * scale_f16 ) -> fp32
//   x:         [8192*64*128/2] int32, each element holds ONE packed byte (2 nibbles)
//   scale:     [8192*64] fp16 (one per 128-element group)
//   zp_packed: [8192*64/2] int32, each element holds ONE packed byte (2 zero points)
//   out:       [8192*8192] fp32
//
// Memory-bound: ~392 MB total traffic -> ~17 us floor at 23.3 TB/s.
// Strategy: 8 outputs per lane; b128 NT load of packed words, 2x b128 NT stores.
// One wave32 covers exactly two quant groups; scale/zp uniform per half-wave.

typedef float vf4 __attribute__((ext_vector_type(4)));
typedef int   vi4 __attribute__((ext_vector_type(4)));

static constexpr int TOTAL_OUT   = 8192 * 8192;       // 67,108,864 fp32
static constexpr int OUT_PER_THR = 8;                 // 2 nibbles x 4 packed words
static constexpr int N_THREADS   = TOTAL_OUT / OUT_PER_THR; // 8,388,608
static constexpr int BLOCK       = 256;               // 8 waves

__global__ __launch_bounds__(BLOCK)
void int4_asym_dequant_kernel(const int* __restrict__ x,
                              const _Float16* __restrict__ scale,
                              const int* __restrict__ zp_packed,
                              float* __restrict__ out)
{
    const int tid = blockIdx.x * BLOCK + threadIdx.x;
    if (tid >= N_THREADS) return;

    const int obase = tid * OUT_PER_THR;   // < 2^27, int math is safe
    const int xbase = tid * 4;             // 4 packed int32 words (16B aligned)
    const int g     = obase >> 7;          // quant group = flat index / 128

    // Streamed, read-once packed weights: 128-bit non-temporal load.
    const vi4 w = __builtin_nontemporal_load(
        reinterpret_cast<const vi4*>(x + xbase));

    // Per-group scale (fp16) and zero point nibble. 16 lanes of the wave share
    // each of these -> L0 broadcast; regular (temporal) loads are correct here.
    const _Float16 s   = scale[g];
    const int      zpb = zp_packed[g >> 1];
    const int      zp  = (g & 1) ? ((zpb >> 4) & 15) : (zpb & 15);

    // Dequantize 8 elements. The nibble-minus-zp difference is an exact small
    // integer; convert to fp16 and multiply IN fp16 to match the reference's
    // rounding, then widen to fp32.
    float f[OUT_PER_THR];
#pragma unroll
    for (int k = 0; k < 4; ++k) {
        const int b  = w[k];               // one packed byte per int32 word
        const int lo = b & 15;
        const int hi = (b >> 4) & 15;
        const _Float16 dlo = (_Float16)(short)(lo - zp);
        const _Float16 dhi = (_Float16)(short)(hi - zp);
        f[2 * k + 0] = (float)(_Float16)(dlo * s);
        f[2 * k + 1] = (float)(_Float16)(dhi * s);
    }

    // 256 MB write-once output: two 128-bit non-temporal stores (TH=NT) so the
    // stream does not evict useful L2 lines.
    vf4 o0 = { f[0], f[1], f[2], f[3] };
    vf4 o1 = { f[4], f[5], f[6], f[7] };
    __builtin_nontemporal_store(o0, reinterpret_cast<vf4*>(out + obase));
    __builtin_nontemporal_store(o1, reinterpret_cast<vf4*>(out + obase + 4));
}

extern "C" void kernel_launch(void* const* d_in, const int* in_sizes, int n_in,
                              void* d_out, int out_size, void* d_ws, size_t ws_size,
                              hipStream_t stream)
{
    const int*      x     = reinterpret_cast<const int*>(d_in[0]);
    const _Float16* scale = reinterpret_cast<const _Float16*>(d_in[1]);
    const int*      zp    = reinterpret_cast<const int*>(d_in[2]);
    float*          out   = reinterpret_cast<float*>(d_out);

    const int grid = (N_THREADS + BLOCK - 1) / BLOCK;   // 32,768 blocks
    int4_asym_dequant_kernel<<<grid, BLOCK, 0, stream>>>(x, scale, zp, out);
}